// CannyEdgeDetector_80874234184300
// MI455X (gfx1250) — compile-verified
//
#include <hip/hip_runtime.h>
#include <math.h>

// ---------------------------------------------------------------------------
// Fused Canny edge detector for MI455X (gfx1250, wave32).
//  - tiles: 32x32 output pixels per 288-thread (9-wave) workgroup
//  - raw tile staged to LDS via global_load_async_to_lds_b128 (ASYNCcnt path)
//  - separable 5-tap Gaussian blur executed on the matrix pipe with
//    V_WMMA_F32_16X16X4_F32 (banded 16x20 x 20x16 products, K in 4-chunks)
//  - sobel / magnitude / atan2 orientation / NMS / thresholds on VALU
// ---------------------------------------------------------------------------

typedef __attribute__((ext_vector_type(2))) float v2f;
typedef __attribute__((ext_vector_type(8))) float v8f;

constexpr int IMG_H = 1024;
constexpr int IMG_W = 1024;
constexpr int HW    = IMG_H * IMG_W;
constexpr int TILE  = 32;
constexpr int NTHR  = 288;          // 9 waves of 32

// unnormalized gaussian(5, std=1) taps, as in the reference
__device__ __forceinline__ float gsel(int d) {
    // returns g[d] for d in [0,4], else 0
    float r = 0.0f;
    r = (d == 0 || d == 4) ? 0.13533528f : r;
    r = (d == 1 || d == 3) ? 0.60653066f : r;
    r = (d == 2)           ? 1.0f        : r;
    return r;
}

__device__ __forceinline__ int nib(unsigned packed, int k) {
    int v = (packed >> (4 * k)) & 0xF;
    return v > 7 ? v - 16 : v;
}

__device__ __forceinline__ void async_load_b128(unsigned ldsOff, const float* base,
                                                unsigned byteOff) {
    // GVS mode: mem = SGPR64 base + VGPR32 offset; dest = LDS byte address VGPR
    asm volatile("global_load_async_to_lds_b128 %0, %1, %2"
                 :: "v"(ldsOff), "v"(byteOff), "s"(base)
                 : "memory");
}

__device__ __forceinline__ void wait_async0() {
#if __has_builtin(__builtin_amdgcn_s_wait_asynccnt)
    __builtin_amdgcn_s_wait_asynccnt(0);
#else
    asm volatile("s_wait_asynccnt 0" ::: "memory");
#endif
}

__global__ __launch_bounds__(NTHR) void
canny_fused_kernel(const float* __restrict__ img, float* __restrict__ out) {
    // raw tile: image rows [oy-4 .. oy+59], cols [ox-4 .. ox+47] (over-tiled
    // for 16x16 WMMA coverage; surplus rows/cols are dead but stay in L2)
    __shared__ __align__(16) float sRaw [64][56];   // stride 56 -> 16B aligned rows
    __shared__ __align__(16) float sHb  [64][49];   // horizontal blur, origin (oy-4, ox-2)
    __shared__ __align__(16) float sBlur[48][49];   // full blur,      origin (oy-2, ox-2)
    __shared__ float sMag[34][35];                  // channel-summed |grad|, origin (oy-1, ox-1)
    __shared__ float sGx [34][35];
    __shared__ float sGy [34][35];

    const int tid  = threadIdx.x;
    const int lane = tid & 31;
    const int wv   = tid >> 5;
    const int ox   = blockIdx.x * TILE;
    const int oy   = blockIdx.y * TILE;

    // zero accumulators
    for (int i = tid; i < 34 * 35; i += NTHR) {
        (&sMag[0][0])[i] = 0.0f;
        (&sGx [0][0])[i] = 0.0f;
        (&sGy [0][0])[i] = 0.0f;
    }
    __syncthreads();

    const int M  = lane & 15;   // WMMA row (A/D) or column (B) index
    const int hh = lane >> 4;   // lane half-select per ISA VGPR layout tables

    for (int c = 0; c < 3; ++c) {
        // ---- stage 0: zero-fill then async-stage raw tile (SAME zero pad) ----
        for (int i = tid; i < 64 * 56; i += NTHR) (&sRaw[0][0])[i] = 0.0f;
        __syncthreads();

        const float* base = img + c * HW;
        for (int i = tid; i < 64 * 13; i += NTHR) {      // 13 x b128 chunks/row
            int r  = i / 13, ch = i % 13;
            int iy = oy - 4 + r;
            int ix = ox - 4 + ch * 4;
            if (iy >= 0 && iy < IMG_H && ix >= 0 && ix + 3 < IMG_W) {
                unsigned ldsOff  = (unsigned)(size_t)&sRaw[r][ch * 4];
                unsigned byteOff = (unsigned)((iy * IMG_W + ix) * 4);
                async_load_b128(ldsOff, base, byteOff);
            }
        }
        wait_async0();
        __syncthreads();

        // ---- stage 1: horizontal blur via WMMA: hb = raw(16x20) * W(20x16) ----
        // hb[y][x] = sum_d raw[y][x+d] * g[d]
        for (int t = wv; t < 12; t += 9) {               // 4x3 tiles of 16x16
            int ty0 = (t / 3) * 16, tx0 = (t % 3) * 16;
            v8f acc = {};
#pragma unroll
            for (int k = 0; k < 5; ++k) {
                v2f a, b;
                // A (raw) 16x4 layout: lane<16 -> K=0,1 ; lane>=16 -> K=2,3
                a.x = sRaw[ty0 + M][tx0 + 4 * k + 2 * hh];
                a.y = sRaw[ty0 + M][tx0 + 4 * k + 2 * hh + 1];
                // B (band matrix) 4x16: row p = 4k + r + 2*hh, col n = M
                b.x = gsel((4 * k + 0 + 2 * hh) - M);
                b.y = gsel((4 * k + 1 + 2 * hh) - M);
                acc = __builtin_amdgcn_wmma_f32_16x16x4_f32(
                    false, a, false, b, (short)0, acc, false, false);
            }
#pragma unroll
            for (int r = 0; r < 8; ++r)                  // D: VGPR r -> row r+8*hh
                sHb[ty0 + r + 8 * hh][tx0 + M] = acc[r];
        }
        __syncthreads();

        // ---- stage 2: vertical blur via WMMA: blur = G(16x20) * hb(20x16) ----
        // blur[y][x] = sum_d g[d] * hb[y+d][x]
        {
            int ty0 = (wv / 3) * 16, tx0 = (wv % 3) * 16;   // 3x3 tiles, 9 waves
            v8f acc = {};
#pragma unroll
            for (int k = 0; k < 5; ++k) {
                v2f a, b;
                // A (band) 16x4: row m = M, col p = 4k + 2*hh + {0,1}
                a.x = gsel((4 * k + 2 * hh)     - M);
                a.y = gsel((4 * k + 2 * hh + 1) - M);
                // B (hb) 4x16: row p = 4k + r + 2*hh, col n = M
                b.x = sHb[ty0 + 4 * k + 0 + 2 * hh][tx0 + M];
                b.y = sHb[ty0 + 4 * k + 1 + 2 * hh][tx0 + M];
                acc = __builtin_amdgcn_wmma_f32_16x16x4_f32(
                    false, a, false, b, (short)0, acc, false, false);
            }
#pragma unroll
            for (int r = 0; r < 8; ++r)
                sBlur[ty0 + r + 8 * hh][tx0 + M] = acc[r];
        }
        __syncthreads();

        // ---- stage 3: emit blurred channel + sobel/magnitude accumulation ----
        float* outBlur = out + c * HW;
        for (int i = tid; i < TILE * TILE; i += NTHR) {
            int y = i >> 5, x = i & 31;
            outBlur[(oy + y) * IMG_W + ox + x] = sBlur[y + 2][x + 2];
        }
        for (int i = tid; i < 34 * 34; i += NTHR) {
            int yy = i / 34, xx = i % 34;
            int by = yy + 1, bx = xx + 1;                // blur-array coords
            float b00 = sBlur[by - 1][bx - 1], b01 = sBlur[by - 1][bx], b02 = sBlur[by - 1][bx + 1];
            float b10 = sBlur[by    ][bx - 1],                           b12 = sBlur[by    ][bx + 1];
            float b20 = sBlur[by + 1][bx - 1], b21 = sBlur[by + 1][bx], b22 = sBlur[by + 1][bx + 1];
            float gx = (b00 - b02) + 2.0f * (b10 - b12) + (b20 - b22);
            float gy = (b00 + 2.0f * b01 + b02) - (b20 + 2.0f * b21 + b22);
            sGx [yy][xx] += gx;
            sGy [yy][xx] += gy;
            sMag[yy][xx] += sqrtf(gx * gx + gy * gy);
        }
        __syncthreads();
    }

    // ---- stage 4: orientation, NMS, thresholds ----
    float* oMag   = out + 3 * HW;
    float* oOri   = out + 4 * HW;
    float* oThin  = out + 5 * HW;
    float* oThr   = out + 6 * HW;
    float* oEarly = out + 7 * HW;

    const unsigned DXP = 0x10FFF011u;  // dx per direction k (signed nibbles)
    const unsigned DYP = 0xFFF01110u;  // dy per direction k

    for (int i = tid; i < TILE * TILE; i += NTHR) {
        int y = i >> 5, x = i & 31;
        int yy = y + 1, xx = x + 1;
        float mag = sMag[yy][xx];
        float deg = atan2f(sGy[yy][xx], sGx[yy][xx]) * (180.0f / 3.14159f) + 180.0f;
        float ori = rintf(deg * (1.0f / 45.0f)) * 45.0f;
        int q  = (int)(ori * (1.0f / 45.0f));
        int ip = q & 7;
        int in_ = (ip + 4) & 7;
        float pos = mag - sMag[yy + nib(DYP, ip )][xx + nib(DXP, ip )];
        float neg = mag - sMag[yy + nib(DYP, in_)][xx + nib(DXP, in_)];
        float thin = (fminf(pos, neg) > 0.0f) ? mag : 0.0f;
        int o = (oy + y) * IMG_W + ox + x;
        oMag  [o] = mag;
        oOri  [o] = ori;
        oThin [o] = thin;
        oThr  [o] = (thin < 10.0f) ? 0.0f : thin;
        oEarly[o] = (mag  < 10.0f) ? 0.0f : mag;
    }
}

extern "C" void kernel_launch(void* const* d_in, const int* in_sizes, int n_in,
                              void* d_out, int out_size, void* d_ws, size_t ws_size,
                              hipStream_t stream) {
    (void)in_sizes; (void)n_in; (void)out_size; (void)d_ws; (void)ws_size;
    const float* img = (const float*)d_in[0];
    float*       out = (float*)d_out;
    dim3 grid(IMG_W / TILE, IMG_H / TILE);   // 32 x 32 workgroups
    canny_fused_kernel<<<grid, NTHR, 0, stream>>>(img, out);
}